// GraphSAINT_18992345383140
// MI455X (gfx1250) — compile-verified
//
#include <hip/hip_runtime.h>

// N = 100000 (divisible by 16), E = 1.6M, D = 128, DOUT = 47
#define D     128
#define DOUT  47
#define TILE_M 16
#define LDA   260   // padded LDS stride for the 256-wide A tile (260 % 64 == 4 -> no bank conflicts)
#define LDA2  132   // padded LDS stride for the 128-wide A tile

typedef float v2f __attribute__((ext_vector_type(2)));
typedef float v8f __attribute__((ext_vector_type(8)));

__device__ __forceinline__ v8f wmma4(v2f a, v2f b, v8f c) {
  // full-precision f32 WMMA: D = A(16x4) * B(4x16) + C(16x16)
  return __builtin_amdgcn_wmma_f32_16x16x4_f32(false, a, false, b, (short)0, c,
                                               false, false);
}

// ---------------------------------------------------------------------------
// Edge scatter: agg[dst,:] += feat[src,:]; optionally cnt[dst] += 1.
// One 32-lane wave per edge, each lane moves a float4 (hardware f32 atomics,
// all L2-resident since the 51MB feature matrix fits in the 192MB L2).
// ---------------------------------------------------------------------------
__global__ void __launch_bounds__(256) scatter_kernel(
    const float* __restrict__ feat, const int* __restrict__ src,
    const int* __restrict__ dst, float* __restrict__ agg,
    float* __restrict__ cnt, int nE, int doCnt) {
  int gid  = blockIdx.x * 256 + threadIdx.x;
  int e    = gid >> 5;
  int lane = gid & 31;
  if (e >= nE) return;
  int s = src[e];
  int d = dst[e];
  const float4 v = ((const float4*)feat)[(size_t)s * (D / 4) + lane];
  float* p = agg + (size_t)d * D + lane * 4;
  unsafeAtomicAdd(p + 0, v.x);   // global_atomic_add_f32
  unsafeAtomicAdd(p + 1, v.y);
  unsafeAtomicAdd(p + 2, v.z);
  unsafeAtomicAdd(p + 3, v.w);
  if (doCnt && lane == 0) unsafeAtomicAdd(cnt + d, 1.0f);
}

// ---------------------------------------------------------------------------
// Fused SAGE layer: out = act( (aggsum/max(cnt,1)) @ Wl + bias + feat @ Wr )
// Block = 16 nodes x 128 outputs; 8 waves, each owns a 16x16 D-tile.
// A-tile (16 x 256 = [mean-agg | self-feat]) staged in LDS; single K=256
// WMMA loop split into the Wl and Wr passes.
// ---------------------------------------------------------------------------
__global__ void __launch_bounds__(256) sage_gemm_kernel(
    const float* __restrict__ feat, const float* __restrict__ aggsum,
    const float* __restrict__ cnt, const float* __restrict__ Wl,
    const float* __restrict__ Wr, const float* __restrict__ bias,
    float* __restrict__ out, int relu) {
  __shared__ float As[TILE_M * LDA];
  __shared__ float invc[TILE_M];
  const int tid = threadIdx.x;
  const int nb  = blockIdx.x * TILE_M;

  if (tid < TILE_M) invc[tid] = 1.0f / fmaxf(cnt[nb + tid], 1.0f);
  __syncthreads();

  // cooperative A-tile load: col<128 -> mean aggregate, else self features
  for (int i = tid; i < TILE_M * 256; i += 256) {
    int row = i >> 8;
    int col = i & 255;
    float v = (col < D) ? aggsum[(size_t)(nb + row) * D + col] * invc[row]
                        : feat[(size_t)(nb + row) * D + (col - D)];
    As[row * LDA + col] = v;
  }
  __syncthreads();

  const int wave = tid >> 5;
  const int lane = tid & 31;
  const int col0 = wave * 16;
  const int mrow = lane & 15;         // A-row for lanes, also B/D column index
  const int koff = (lane >> 4) * 2;   // K sub-offset per ISA A/B layout
  const float* a_base = &As[mrow * LDA];
  v8f acc = {};

  // pass 1: mean-agg against Wl
  for (int k = 0; k < D; k += 4) {
    v2f a; a.x = a_base[k + koff]; a.y = a_base[k + koff + 1];
    const float* wp = Wl + (size_t)(k + koff) * D + col0 + mrow;
    v2f b; b.x = wp[0]; b.y = wp[D];
    acc = wmma4(a, b, acc);
  }
  // pass 2: self features against Wr
  for (int k = 0; k < D; k += 4) {
    v2f a; a.x = a_base[D + k + koff]; a.y = a_base[D + k + koff + 1];
    const float* wp = Wr + (size_t)(k + koff) * D + col0 + mrow;
    v2f b; b.x = wp[0]; b.y = wp[D];
    acc = wmma4(a, b, acc);
  }

  const float bc   = bias[col0 + mrow];
  const int  rbase = (lane >> 4) * 8;      // C/D layout: M = v (+8 for upper half)
  for (int v = 0; v < 8; ++v) {
    float val = acc[v] + bc;
    if (relu) val = fmaxf(val, 0.0f);
    out[(size_t)(nb + rbase + v) * D + col0 + mrow] = val;
  }
}

// ---------------------------------------------------------------------------
// Final projection: out = h @ Wout + bout, 128 -> 47 (3 waves, cols >= 47
// masked only in loads/stores; EXEC is all-ones for every WMMA).
// ---------------------------------------------------------------------------
__global__ void __launch_bounds__(96) out_gemm_kernel(
    const float* __restrict__ h, const float* __restrict__ Wout,
    const float* __restrict__ bias, float* __restrict__ out) {
  __shared__ float As[TILE_M * LDA2];
  const int tid = threadIdx.x;
  const int nb  = blockIdx.x * TILE_M;

  for (int i = tid; i < TILE_M * D; i += 96) {
    int row = i >> 7;
    int col = i & 127;
    As[row * LDA2 + col] = h[(size_t)(nb + row) * D + col];
  }
  __syncthreads();

  const int wave = tid >> 5;
  const int lane = tid & 31;
  const int col0 = wave * 16;
  const int mrow = lane & 15;
  const int koff = (lane >> 4) * 2;
  const int col  = col0 + mrow;
  const float* a_base = &As[mrow * LDA2];
  v8f acc = {};

  for (int k = 0; k < D; k += 4) {
    v2f a; a.x = a_base[k + koff]; a.y = a_base[k + koff + 1];
    v2f b;
    b.x = (col < DOUT) ? Wout[(size_t)(k + koff) * DOUT + col] : 0.0f;
    b.y = (col < DOUT) ? Wout[(size_t)(k + koff + 1) * DOUT + col] : 0.0f;
    acc = wmma4(a, b, acc);
  }

  if (col < DOUT) {
    const float bc   = bias[col];
    const int  rbase = (lane >> 4) * 8;
    for (int v = 0; v < 8; ++v)
      out[(size_t)(nb + rbase + v) * DOUT + col] = acc[v] + bc;
  }
}

// ---------------------------------------------------------------------------
extern "C" void kernel_launch(void* const* d_in, const int* in_sizes, int n_in,
                              void* d_out, int out_size, void* d_ws, size_t ws_size,
                              hipStream_t stream) {
  const float* x    = (const float*)d_in[0];
  const int*   edge = (const int*)d_in[1];   // [2, E] (src row, dst row)
  // d_in[2] = relations (unused by the reference forward)
  const float* Wl1  = (const float*)d_in[3];
  const float* Wr1  = (const float*)d_in[4];
  const float* b1   = (const float*)d_in[5];
  const float* Wl2  = (const float*)d_in[6];
  const float* Wr2  = (const float*)d_in[7];
  const float* b2   = (const float*)d_in[8];
  const float* Wout = (const float*)d_in[9];
  const float* bout = (const float*)d_in[10];

  const int N = in_sizes[0] / D;   // 100000 (multiple of 16)
  const int E = in_sizes[1] / 2;   // 1600000
  const int* src = edge;
  const int* dst = edge + E;

  // workspace: [cnt N][agg N*128][h1 N*128]  (~103 MB); h2 reuses the agg
  // buffer in-place (each GEMM block reads only the rows it writes).
  float* ws   = (float*)d_ws;
  size_t aOff = ((size_t)N + 255) & ~(size_t)255;
  float* cnt  = ws;
  float* agg  = ws + aOff;
  float* h1   = agg + (size_t)N * D;

  const int scatterBlocks = (E * 32 + 255) / 256;
  const int mTiles        = N / TILE_M;

  // ---- layer 1 ----
  hipMemsetAsync(cnt, 0, sizeof(float) * (size_t)N, stream);
  hipMemsetAsync(agg, 0, sizeof(float) * (size_t)N * D, stream);
  scatter_kernel<<<scatterBlocks, 256, 0, stream>>>(x, src, dst, agg, cnt, E, 1);
  sage_gemm_kernel<<<mTiles, 256, 0, stream>>>(x, agg, cnt, Wl1, Wr1, b1, h1, 1);

  // ---- layer 2 (degree counts are identical; reuse cnt) ----
  hipMemsetAsync(agg, 0, sizeof(float) * (size_t)N * D, stream);
  scatter_kernel<<<scatterBlocks, 256, 0, stream>>>(h1, src, dst, agg, cnt, E, 0);
  sage_gemm_kernel<<<mTiles, 256, 0, stream>>>(h1, agg, cnt, Wl2, Wr2, b2, agg, 0);

  // ---- output projection ----
  out_gemm_kernel<<<mTiles, 96, 0, stream>>>(agg, Wout, bout, (float*)d_out);
}